// LogicConstraintLoss_70257075028115
// MI455X (gfx1250) — compile-verified
//
#include <hip/hip_runtime.h>

// Problem constants (match reference)
#define BB 2
#define NN 320
#define RR 6
#define KK 16
#define NBLK 256   // pair-kernel blocks; finalize reduces exactly 256 partials
#define NTHR 256

typedef float v2f __attribute__((ext_vector_type(2)));
typedef float v8f __attribute__((ext_vector_type(8)));

// Exact FP32 sum of 256 LDS floats using V_WMMA_F32_16X16X4_F32.
// A = ones(16x4); B chunks hold 64 values as B[k,n] = v[ch*64 + k*16 + n]
// per the ISA 4x16 f32 B-matrix lane layout. After 4 accumulating WMMAs,
// D[0,n] (VGPR0, lanes 0..15) holds the 16 column sums; a 16-lane xor-fold
// finishes. Must be called by all lanes with EXEC all-ones (uniform path).
__device__ __forceinline__ float wmma_reduce256(const float* __restrict__ v) {
  const unsigned lane = threadIdx.x & 31u;
  v2f a; a[0] = 1.0f; a[1] = 1.0f;                // ones matrix (both K pairs)
  v8f c = {0.f, 0.f, 0.f, 0.f, 0.f, 0.f, 0.f, 0.f};
  const unsigned i0 = (lane < 16u) ? lane : (lane + 16u);  // B vgpr0 source
  const unsigned i1 = i0 + 16u;                            // B vgpr1 source
#pragma unroll
  for (int ch = 0; ch < 4; ++ch) {
    v2f b; b[0] = v[ch * 64 + i0]; b[1] = v[ch * 64 + i1];
    c = __builtin_amdgcn_wmma_f32_16x16x4_f32(false, a, false, b,
                                              (short)0, c, false, false);
  }
  float s = c[0];                 // lanes 0..15: column sums (lanes 16..31 dup)
  s += __shfl_xor(s, 8, 32);
  s += __shfl_xor(s, 4, 32);
  s += __shfl_xor(s, 2, 32);
  s += __shfl_xor(s, 1, 32);
  return s;                       // total of all 256 values (all lanes valid)
}

// Streaming pass over all (b,i,j) pairs: sym / excl0 / excl1 / pm-count.
__global__ void pair_kernel(const float* __restrict__ rp,
                            const unsigned char* __restrict__ mask,
                            float* __restrict__ blk_part) {
  __shared__ float red[4][NTHR];
  float sym = 0.f, e0 = 0.f, e1 = 0.f, cnt = 0.f;
  const int total = BB * NN * NN;
  for (int idx = blockIdx.x * NTHR + threadIdx.x; idx < total;
       idx += NBLK * NTHR) {
    int b   = idx / (NN * NN);
    int rem = idx - b * (NN * NN);
    int i   = rem / NN;
    int j   = rem - i * NN;
    if (i != j && mask[b * NN + i] && mask[b * NN + j]) {
      const float* p = rp + (size_t)idx * RR;                       // [b,i,j,:]
      const float* q = rp + (size_t)((b * NN + j) * NN + i) * RR;   // [b,j,i,:]
      float p0 = p[0], p1 = p[1], p2 = p[2], p3 = p[3], p4 = p[4], p5 = p[5];
      float q4 = q[4], q5 = q[5];
      sym += fabsf(p4 - q4) + fabsf(p5 - q5);
      e0  += p0 * p1;
      e1  += p2 * p3;
      cnt += 1.0f;
    }
  }
  red[0][threadIdx.x] = sym;
  red[1][threadIdx.x] = e0;
  red[2][threadIdx.x] = e1;
  red[3][threadIdx.x] = cnt;
  __syncthreads();
  // Uniform: every wave runs the WMMA reduction with full EXEC.
  float t0 = wmma_reduce256(red[0]);
  float t1 = wmma_reduce256(red[1]);
  float t2 = wmma_reduce256(red[2]);
  float t3 = wmma_reduce256(red[3]);
  if (threadIdx.x == 0) {
    blk_part[0 * NBLK + blockIdx.x] = t0;
    blk_part[1 * NBLK + blockIdx.x] = t1;
    blk_part[2 * NBLK + blockIdx.x] = t2;
    blk_part[3 * NBLK + blockIdx.x] = t3;
  }
}

// Single block: reduce block partials (WMMA again), compute the j==0
// transitivity term with knn de-dup, emit the 3 outputs deterministically.
__global__ void finalize_kernel(const float* __restrict__ rp,
                                const unsigned char* __restrict__ mask,
                                const int* __restrict__ knn,
                                const float* __restrict__ blk_part,
                                float* __restrict__ out) {
  __shared__ float red[NTHR];
  __shared__ int   imc[NTHR];
  const int tid = threadIdx.x;

  float totals[4];
#pragma unroll
  for (int q = 0; q < 4; ++q) {
    red[tid] = blk_part[q * NBLK + tid];
    __syncthreads();
    totals[q] = wmma_reduce256(red);
    __syncthreads();
  }

  // ---- transitivity (j forced to 0 by the reference's scatter quirk) ----
  float tt = 0.f;
  int   mc = 0;
  for (int e = tid; e < BB * NN; e += NTHR) {
    int b = e / NN;
    int i = e - b * NN;
    if (i == 0) continue;
    if (!mask[b * NN + 0] || !mask[b * NN + i]) continue;
    const int* kr = knn + (b * NN + i) * KK;
    for (int t = 0; t < KK; ++t) {
      int k = kr[t];
      bool dup = false;
      for (int t2 = 0; t2 < t; ++t2)
        if (kr[t2] == k) { dup = true; break; }
      if (dup) continue;                    // sampled_k marks existence once
      if (k == 0 || k == i) continue;       // j!=k (j==0) and i!=k and pm[k,0]
      if (!mask[b * NN + k]) continue;
      mc += 1;
      const float* r_i0 = rp + (size_t)((b * NN + i) * NN + 0) * RR;
      const float* r_0k = rp + (size_t)((b * NN + 0) * NN + k) * RR;
      const float* r_ik = rp + (size_t)((b * NN + i) * NN + k) * RR;
#pragma unroll
      for (int rr = 0; rr < 2; ++rr) {
        const int r = (rr == 0) ? 0 : 2;    // TRANSITIVE = (before, contains)
        float prem = r_i0[r] + r_0k[r] - 1.0f;
        prem = prem > 0.f ? prem : 0.f;     // Lukasiewicz t-norm clip
        float viol = prem - r_ik[r];
        viol = viol > 0.f ? viol : 0.f;     // relu
        tt += viol;
      }
    }
  }
  red[tid] = tt;
  imc[tid] = mc;
  __syncthreads();
  float tt_total = wmma_reduce256(red);     // exact f32 WMMA reduce again
  __syncthreads();
  for (int s = NTHR / 2; s > 0; s >>= 1) {  // deterministic int tree reduce
    if (tid < s) imc[tid] += imc[tid + s];
    __syncthreads();
  }

  if (tid == 0) {
    float pm_cnt = totals[3];               // integer-valued, exact in f32
    float denom  = pm_cnt > 1.f ? pm_cnt : 1.f;
    float sym    = totals[0] / denom;
    float excl   = (totals[1] / denom + totals[2] / denom) * 0.5f;
    int   ms     = imc[0] > 1 ? imc[0] : 1;
    float trans  = tt_total / (float)(2 * ms);   // count = 2*max(masksum,1)
    out[0] = sym;
    out[1] = trans;
    out[2] = excl;
  }
}

extern "C" void kernel_launch(void* const* d_in, const int* in_sizes, int n_in,
                              void* d_out, int out_size, void* d_ws,
                              size_t ws_size, hipStream_t stream) {
  (void)in_sizes; (void)n_in; (void)out_size; (void)ws_size;
  const float*         rp   = (const float*)d_in[0];         // [B,N,N,R] f32
  const unsigned char* mask = (const unsigned char*)d_in[1]; // [B,N] bool
  const int*           knn  = (const int*)d_in[2];           // [B,N,K] i32
  float*               out  = (float*)d_out;                 // 3 x f32
  float*               blk  = (float*)d_ws;                  // 4*NBLK floats

  pair_kernel<<<NBLK, NTHR, 0, stream>>>(rp, mask, blk);
  finalize_kernel<<<1, NTHR, 0, stream>>>(rp, mask, knn, blk, out);
}